// GNN_6305011991202
// MI455X (gfx1250) — compile-verified
//
#include <hip/hip_runtime.h>
#include <hip/hip_bf16.h>

// ---------------------------------------------------------------------------
// GraphSAGE 3-layer forward for MI455X (gfx1250), wave32, WMMA fp32 path.
//   N=50000 nodes, E=640000 edges, D_IN=HIDDEN=128, OUT=64
//   layer: h = relu( mean_agg(x)@Wl.T + bl + x@Wr.T )
//   out   = h2@W3.T + b3
// ---------------------------------------------------------------------------

typedef __attribute__((ext_vector_type(2))) float v2f;
typedef __attribute__((ext_vector_type(8))) float v8f;

#define N_NODES 50000
#define N_EDGES 640000
#define D_FEAT 128   // D_IN == HIDDEN
#define D_OUT 64

// --- degree count: one fp32 atomic per edge --------------------------------
__global__ void count_kernel(const int* __restrict__ dst, float* __restrict__ cnt,
                             int nEdges) {
    int e = blockIdx.x * blockDim.x + threadIdx.x;
    if (e < nEdges) atomicAdd(&cnt[dst[e]], 1.0f);
}

// --- segment-sum scatter: 32 threads per edge, float4 per thread -----------
__global__ void scatter_kernel(const float* __restrict__ feat,
                               const int* __restrict__ src,
                               const int* __restrict__ dst,
                               float* __restrict__ agg, int nEdges) {
    int t = blockIdx.x * blockDim.x + threadIdx.x;
    int e = t >> 5;          // edge index
    int g = t & 31;          // feature group (4 floats each -> 128)
    if (e >= nEdges) return;
    int s = src[e];
    int d = dst[e];
    const float4 v = *(const float4*)&feat[(size_t)s * D_FEAT + g * 4];
    float* ap = &agg[(size_t)d * D_FEAT + g * 4];
    atomicAdd(ap + 0, v.x);
    atomicAdd(ap + 1, v.y);
    atomicAdd(ap + 2, v.z);
    atomicAdd(ap + 3, v.w);
}

// --- fused SAGE GEMM using V_WMMA_F32_16X16X4_F32 --------------------------
// Block = one 16-row tile of nodes. Wave y = one 16-col tile of the output.
//   C[16x16] = mean_tile @ Wl.T + x_tile @ Wr.T   (each term optional)
//   out = (relu?) (C + bias)
// A tiles staged in LDS (stride 132 floats: 16 rows land in distinct banks).
// Weight fragments read as aligned float2 from global (L2/WGP$-resident).
__global__ void sage_gemm(const float* __restrict__ agg,   // may be null
                          const float* __restrict__ cnt,   // used iff agg
                          const float* __restrict__ xin,   // may be null
                          const float* __restrict__ Wl,    // [outDim, K], used iff agg
                          const float* __restrict__ bias,  // [outDim]
                          const float* __restrict__ Wr,    // [outDim, K], used iff xin
                          float* __restrict__ out,         // [N, outDim]
                          int K, int outDim, int doRelu) {
    __shared__ float sMean[16 * 132];
    __shared__ float sX[16 * 132];
    __shared__ float sRc[16];

    const int tid = threadIdx.y * 32 + threadIdx.x;
    const int nthreads = blockDim.y * 32;
    const int m0 = blockIdx.x * 16;

    // reciprocal of clamped degree, one per tile row
    if (agg != nullptr && tid < 16) {
        sRc[tid] = 1.0f / fmaxf(cnt[m0 + tid], 1.0f);
    }
    __syncthreads();

    // stage A tiles into LDS (padded stride 132)
    for (int i = tid; i < 16 * 128; i += nthreads) {
        int r = i >> 7;
        int c = i & 127;
        size_t gidx = (size_t)(m0 + r) * K + c;
        if (agg != nullptr) sMean[r * 132 + c] = agg[gidx] * sRc[r];
        if (xin != nullptr) sX[r * 132 + c] = xin[gidx];
    }
    __syncthreads();

    const int L = threadIdx.x;        // lane within wave
    const int half = L >> 4;          // 0: K=k0..k0+1, 1: K=k0+2..k0+3
    const int l16 = L & 15;           // A: row, B: col, C/D: col
    const int n16 = threadIdx.y * 16; // output col tile base

    v8f c = {};

    // term 1: mean @ Wl.T
    if (agg != nullptr) {
        const float* wbase = Wl + (size_t)(n16 + l16) * K;
        for (int kt = 0; kt < K / 4; ++kt) {
            int k = kt * 4 + 2 * half;
            v2f a = *(const v2f*)&sMean[l16 * 132 + k];
            v2f b = *(const v2f*)&wbase[k];
            c = __builtin_amdgcn_wmma_f32_16x16x4_f32(
                false, a, false, b, (short)0, c, false, false);
        }
    }
    // term 2: x @ Wr.T
    if (xin != nullptr) {
        const float* wbase = Wr + (size_t)(n16 + l16) * K;
        for (int kt = 0; kt < K / 4; ++kt) {
            int k = kt * 4 + 2 * half;
            v2f a = *(const v2f*)&sX[l16 * 132 + k];
            v2f b = *(const v2f*)&wbase[k];
            c = __builtin_amdgcn_wmma_f32_16x16x4_f32(
                false, a, false, b, (short)0, c, false, false);
        }
    }

    // bias + relu + store. C/D layout: VGPR j -> row (half? j+8 : j), col l16.
    float bv = bias[n16 + l16];
#pragma unroll
    for (int j = 0; j < 8; ++j) {
        float v = c[j] + bv;
        if (doRelu) v = fmaxf(v, 0.0f);
        int row = m0 + (half ? j + 8 : j);
        out[(size_t)row * outDim + n16 + l16] = v;
    }
}

extern "C" void kernel_launch(void* const* d_in, const int* in_sizes, int n_in,
                              void* d_out, int out_size, void* d_ws, size_t ws_size,
                              hipStream_t stream) {
    const float* x = (const float*)d_in[0];
    const int* edge_index = (const int*)d_in[1];     // [2, E] int32
    const float* W1_l = (const float*)d_in[2];
    const float* b1_l = (const float*)d_in[3];
    const float* W1_r = (const float*)d_in[4];
    const float* W2_l = (const float*)d_in[5];
    const float* b2_l = (const float*)d_in[6];
    const float* W2_r = (const float*)d_in[7];
    const float* W3 = (const float*)d_in[8];
    const float* b3 = (const float*)d_in[9];
    float* out = (float*)d_out;

    const int* src = edge_index;
    const int* dst = edge_index + N_EDGES;

    // workspace layout (floats)
    float* ws = (float*)d_ws;
    const size_t szFeat = (size_t)N_NODES * D_FEAT;          // 6.4M floats
    float* agg = ws;                                         // [N, 128]
    float* cnt = agg + szFeat;                               // [N]
    float* h1 = cnt + ((N_NODES + 63) / 64) * 64;            // [N, 128]
    float* h2 = h1 + szFeat;                                 // [N, 128]

    const int rowTiles = N_NODES / 16;                       // 3125 exactly
    dim3 gemmGridH(rowTiles);
    dim3 gemmBlockH(32, D_FEAT / 16);                        // 8 waves
    dim3 gemmBlockO(32, D_OUT / 16);                         // 4 waves

    const int scatterThreads = N_EDGES * 32;
    dim3 scatterGrid((scatterThreads + 255) / 256);
    dim3 countGrid((N_EDGES + 255) / 256);

    // ---- degrees (shared by both layers) ----
    hipMemsetAsync(cnt, 0, (size_t)N_NODES * sizeof(float), stream);
    count_kernel<<<countGrid, 256, 0, stream>>>(dst, cnt, N_EDGES);

    // ---- layer 1 ----
    hipMemsetAsync(agg, 0, szFeat * sizeof(float), stream);
    scatter_kernel<<<scatterGrid, 256, 0, stream>>>(x, src, dst, agg, N_EDGES);
    sage_gemm<<<gemmGridH, gemmBlockH, 0, stream>>>(
        agg, cnt, x, W1_l, b1_l, W1_r, h1, D_FEAT, D_FEAT, 1);

    // ---- layer 2 ----
    hipMemsetAsync(agg, 0, szFeat * sizeof(float), stream);
    scatter_kernel<<<scatterGrid, 256, 0, stream>>>(h1, src, dst, agg, N_EDGES);
    sage_gemm<<<gemmGridH, gemmBlockH, 0, stream>>>(
        agg, cnt, h1, W2_l, b2_l, W2_r, h2, D_FEAT, D_FEAT, 1);

    // ---- layer 3: out = h2 @ W3.T + b3 ----
    sage_gemm<<<gemmGridH, gemmBlockO, 0, stream>>>(
        nullptr, nullptr, h2, nullptr, b3, W3, out, D_FEAT, D_OUT, 0);
}